// PositionalEncoding_56805237457391
// MI455X (gfx1250) — compile-verified
//
#include <hip/hip_runtime.h>
#include <math.h>
#include <stdint.h>

// out[b,c,h,w] = x[b,c,h,w] + pe[c,h]   (sinusoidal positional encoding add)
// Pure HBM-bandwidth-bound streaming kernel: 512 MiB traffic, ~22 us floor
// at 23.3 TB/s. One block per (c,h) so sin/cos is computed once per 32 KiB of
// traffic (and wave-uniform -> scalarized by the compiler); 128-bit accesses;
// non-temporal hints on BOTH streams (512 MiB stream >> 192 MB L2); async
// global->LDS double-buffered pipeline (CDNA5 ASYNCcnt path).

typedef float v4f __attribute__((ext_vector_type(4)));
typedef int   v4i __attribute__((vector_size(4 * sizeof(int))));  // builtin's pointee type

#ifndef __has_builtin
#define __has_builtin(x) 0
#endif

#if __has_builtin(__builtin_amdgcn_global_load_async_to_lds_b128) && \
    __has_builtin(__builtin_amdgcn_s_wait_asynccnt)
#define PE_USE_ASYNC 1
#else
#define PE_USE_ASYNC 0
#endif

#define PE_THREADS 256
#define CPOL_NT 1                       // TH[2:0] = 1 (non-temporal), scope = 0

template <int C, int H, int W, int B>
__global__ __launch_bounds__(PE_THREADS) void pe_add_kernel(
    const float* __restrict__ x, float* __restrict__ out, float k)
{
    static_assert((H & (H - 1)) == 0 && (W & (W - 1)) == 0, "pow2 dims");

    const int t  = threadIdx.x;
    const int ch = blockIdx.x;                 // linear (c,h)
    const int c  = ch >> 8;                    // H = 256
    const int h  = ch & (H - 1);

    // pe[c,h] = sin/cos(h * 10000^(-2*(c/2)/C)) ; k = -2*log2(10000)/C
    const float arg = (float)h * exp2f((float)(c >> 1) * k);
    const float pe  = (c & 1) ? cosf(arg) : sinf(arg);

    constexpr int VPR    = W >> 2;                         // 64 float4 per row
    constexpr int PLANE  = C * H * W;                      // elems per batch image
    constexpr int CHUNKS = (B * VPR) / PE_THREADS;         // 4
    const size_t rowbase = ((size_t)c * H + h) * (size_t)W;

    // per-chunk global element offset for this thread (all shifts/masks)
    auto goff = [&](int i) -> size_t {
        int lin = i * PE_THREADS + t;
        int b   = lin >> 6;                                // / VPR
        int w4  = lin & (VPR - 1);
        return rowbase + (size_t)b * (size_t)PLANE + (size_t)(w4 << 2);
    };

#if PE_USE_ASYNC
    __shared__ v4f lds[2][PE_THREADS];                     // 8 KiB double buffer

    auto issue = [&](int i) {
        v4i* src = (v4i*)(x + goff(i));                    // C-cast drops const
        v4i* dst = (v4i*)&lds[i & 1][t];
        __builtin_amdgcn_global_load_async_to_lds_b128(src, dst,
                                                       /*offset=*/0,
                                                       /*cpol=*/CPOL_NT);
    };

    issue(0);
#pragma unroll
    for (int i = 0; i < CHUNKS; ++i) {
        if (i + 1 < CHUNKS) {
            issue(i + 1);                                  // keep 2 chunks in flight
            __builtin_amdgcn_s_wait_asynccnt(1);           // chunk i landed in LDS
        } else {
            __builtin_amdgcn_s_wait_asynccnt(0);
        }
        asm volatile("" ::: "memory");                     // no reordering across wait

        v4f v = lds[i & 1][t];                             // ds_load_b128, own slot:
        v = v + pe;                                        // no barrier needed
        float* dstp = out + goff(i);
        __builtin_nontemporal_store(v, (v4f*)dstp);        // NT: don't pollute L2
    }
#else
    // Fallback: direct non-temporal 128-bit streaming
#pragma unroll
    for (int i = 0; i < CHUNKS; ++i) {
        size_t off = goff(i);
        v4f v = __builtin_nontemporal_load((const v4f*)(x + off));
        v = v + pe;
        __builtin_nontemporal_store(v, (v4f*)(out + off));
    }
#endif
}

extern "C" void kernel_launch(void* const* d_in, const int* in_sizes, int n_in,
                              void* d_out, int out_size, void* d_ws, size_t ws_size,
                              hipStream_t stream) {
    (void)in_sizes; (void)n_in; (void)d_ws; (void)ws_size; (void)out_size;
    const float* x   = (const float*)d_in[0];
    float*       out = (float*)d_out;

    constexpr int B = 16, C = 64, H = 256, W = 256;
    const float k = -2.0f * log2f(10000.0f) / (float)C;

    pe_add_kernel<C, H, W, B><<<dim3(C * H), dim3(PE_THREADS), 0, stream>>>(x, out, k);
}